// TimeDiscriminator_25890062860996
// MI455X (gfx1250) — compile-verified
//
#include <hip/hip_runtime.h>

#define NH 128  // hidden size fixed by the problem (in_sizes[6] == 128)

typedef __attribute__((ext_vector_type(2))) float v2f;
typedef __attribute__((ext_vector_type(8))) float v8f;

// ---------------------------------------------------------------------------
// 1) Exclusive prefix scan of grid_sizes -> offsets[G+1]   (single block)
// ---------------------------------------------------------------------------
__global__ void td_scan_kernel(const int* __restrict__ sizes, int G,
                               int* __restrict__ offsets) {
  __shared__ int sdata[1024];
  __shared__ int carry_s;
  const int tid = threadIdx.x;
  if (tid == 0) carry_s = 0;
  __syncthreads();
  for (int base = 0; base < G; base += 1024) {
    const int i = base + tid;
    const int v = (i < G) ? sizes[i] : 0;
    sdata[tid] = v;
    __syncthreads();
    for (int off = 1; off < 1024; off <<= 1) {
      int add = (tid >= off) ? sdata[tid - off] : 0;
      __syncthreads();
      sdata[tid] += add;
      __syncthreads();
    }
    const int incl = sdata[tid];
    const int carry = carry_s;
    if (i < G) offsets[i] = carry + incl - v;  // exclusive prefix
    __syncthreads();
    if (tid == 0) carry_s = carry + sdata[1023];
    __syncthreads();
  }
  if (tid == 0) offsets[G] = carry_s;
}

// ---------------------------------------------------------------------------
// 2) seg_ids[p] = g for p in [offsets[g], offsets[g+1])
// ---------------------------------------------------------------------------
__global__ void td_segfill_kernel(const int* __restrict__ offsets, int G,
                                  int* __restrict__ seg_ids) {
  const int g = blockIdx.x * blockDim.x + threadIdx.x;
  if (g >= G) return;
  const int b = offsets[g], e = offsets[g + 1];
  for (int p = b; p < e; ++p) seg_ids[p] = g;
}

// ---------------------------------------------------------------------------
// 3) M[g,:] = mean of embedding_[pos_samples[p],:] over segment g
//    one wave32 per grid; lane covers 4 contiguous floats (b128 loads)
// ---------------------------------------------------------------------------
__global__ void td_gridmean_kernel(const float* __restrict__ emb_,
                                   const int* __restrict__ pos,
                                   const int* __restrict__ offsets, int G,
                                   float* __restrict__ M) {
  const int lane = threadIdx.x & 31;
  const int g = blockIdx.x * 8 + (threadIdx.x >> 5);
  if (g >= G) return;
  const int b = offsets[g], e = offsets[g + 1];
  float ax = 0.f, ay = 0.f, az = 0.f, aw = 0.f;
  for (int p = b; p < e; ++p) {
    const int idx = pos[p];
    const float4 v = *(const float4*)(emb_ + (size_t)idx * NH + (lane << 2));
    ax += v.x; ay += v.y; az += v.z; aw += v.w;
  }
  const float inv = (e > b) ? (1.f / (float)(e - b)) : 0.f;
  float4 o;
  o.x = ax * inv; o.y = ay * inv; o.z = az * inv; o.w = aw * inv;
  *(float4*)(M + (size_t)g * NH + (lane << 2)) = o;
}

// ---------------------------------------------------------------------------
// 4) One-block setup:  T1 = Wk@Wi ; A = Wi^T@T1 ;
//    d = Wi^T (Wk bi) ; q = Wi^T (Wk^T bi) ; s = bi^T Wk bi + bk
//    vecs layout: [0..127]=d, [128..255]=q, [256]=s
// ---------------------------------------------------------------------------
__global__ void td_setup_kernel(const float* __restrict__ Wi,
                                const float* __restrict__ bi,
                                const float* __restrict__ Wk,
                                const float* __restrict__ bk,
                                float* __restrict__ T1,
                                float* __restrict__ Amat,
                                float* __restrict__ vecs) {
  __shared__ float t[NH], t2[NH];
  const int tid = threadIdx.x;  // 256 threads
  for (int idx = tid; idx < NH * NH; idx += blockDim.x) {
    const int d = idx / NH, k = idx % NH;
    float acc = 0.f;
    for (int e = 0; e < NH; ++e) acc += Wk[d * NH + e] * Wi[e * NH + k];
    T1[idx] = acc;
  }
  if (tid < NH) {
    float a = 0.f, b = 0.f;
    for (int e = 0; e < NH; ++e) {
      a += Wk[tid * NH + e] * bi[e];   // (Wk  bi)[tid]
      b += Wk[e * NH + tid] * bi[e];   // (Wk^T bi)[tid]
    }
    t[tid] = a; t2[tid] = b;
  }
  __threadfence();
  __syncthreads();
  for (int idx = tid; idx < NH * NH; idx += blockDim.x) {
    const int j = idx / NH, k = idx % NH;
    float acc = 0.f;
    for (int d = 0; d < NH; ++d) acc += Wi[d * NH + j] * T1[d * NH + k];
    Amat[idx] = acc;  // A[j,k] = (Wi^T Wk Wi)[j,k]
  }
  if (tid < NH) {
    float dv = 0.f, qv = 0.f;
    for (int d = 0; d < NH; ++d) {
      dv += Wi[d * NH + tid] * t[d];
      qv += Wi[d * NH + tid] * t2[d];
    }
    vecs[tid] = dv;
    vecs[NH + tid] = qv;
  }
  if (tid == 0) {
    float s = bk[0];
    for (int d = 0; d < NH; ++d) s += bi[d] * t[d];
    vecs[2 * NH] = s;
  }
}

// ---------------------------------------------------------------------------
// 5) U[G,128] = M @ A^T + d   via V_WMMA_F32_16X16X4_F32
//    one wave per 16x16 output tile; 8 waves/block cover all 8 column tiles
//    A-matrix tile (16x4): lane = M-row, VGPR0/1 = K+0/1 (lanes<16), K+2/3 (>=16)
//    B-matrix tile (4x16): same load shape since B = A^T row-major
// ---------------------------------------------------------------------------
__global__ void td_u_wmma_kernel(const float* __restrict__ M,
                                 const float* __restrict__ Amat,
                                 const float* __restrict__ vecs,
                                 float* __restrict__ U, int G) {
  const int lane = threadIdx.x & 31;
  const int jbase = (threadIdx.x >> 5) * 16;  // 0..112
  const int gbase = blockIdx.x * 16;
  const int lr = lane & 15;
  const int khi = (lane >= 16) ? 2 : 0;

  int grow = gbase + lr;
  if (grow >= G) grow = G - 1;  // clamp reads (G multiple of 16 in practice)
  const float* arow = M + (size_t)grow * NH;
  const float* brow = Amat + (size_t)(jbase + lr) * NH;

  v8f acc = {};
#pragma unroll 8
  for (int k0 = 0; k0 < NH; k0 += 4) {
    const v2f a = *(const v2f*)(arow + k0 + khi);
    const v2f b = *(const v2f*)(brow + k0 + khi);
    acc = __builtin_amdgcn_wmma_f32_16x16x4_f32(
        /*neg_a=*/false, a, /*neg_b=*/false, b,
        /*c_mod=*/(short)0, acc, /*reuse_a=*/false, /*reuse_b=*/false);
  }

  // C/D layout: VGPR r, lanes0-15 -> row gbase+r, lanes16-31 -> row gbase+r+8
  const int gofs = (lane >= 16) ? 8 : 0;
  const float dj = vecs[jbase + lr];
  float* ubase = U + (size_t)(gbase + gofs) * NH + jbase + lr;
  if (gbase + 15 < G) {
    // full tile: unconditional stores, no exec-mask churn
#pragma unroll
    for (int r = 0; r < 8; ++r) {
      ubase[(size_t)r * NH] = acc[r] + dj;
    }
  } else {
    // edge tile (only when G % 16 != 0)
#pragma unroll
    for (int r = 0; r < 8; ++r) {
      const int g = gbase + r + gofs;
      if (g < G) U[(size_t)g * NH + jbase + lr] = acc[r] + dj;
    }
  }
}

// ---------------------------------------------------------------------------
// 6) c[g] = q . M[g] + s     (one wave per grid)
// ---------------------------------------------------------------------------
__global__ void td_c_kernel(const float* __restrict__ M,
                            const float* __restrict__ vecs, int G,
                            float* __restrict__ cg) {
  const int lane = threadIdx.x & 31;
  const int g = blockIdx.x * 8 + (threadIdx.x >> 5);
  if (g >= G) return;
  const float4 q = *(const float4*)(vecs + NH + (lane << 2));
  const float4 m = *(const float4*)(M + (size_t)g * NH + (lane << 2));
  float p = q.x * m.x + q.y * m.y + q.z * m.z + q.w * m.w;
  for (int o = 16; o; o >>= 1) p += __shfl_xor(p, o, 32);
  if (lane == 0) cg[g] = p + vecs[2 * NH];
}

// ---------------------------------------------------------------------------
// 7) logits: out[s] = embedding[idx] . U[g] + c[g]   (one wave per sample)
// ---------------------------------------------------------------------------
__global__ void td_logits_kernel(const float* __restrict__ emb,
                                 const int* __restrict__ pos,
                                 const int* __restrict__ neg,
                                 const int* __restrict__ seg_ids,
                                 const float* __restrict__ U,
                                 const float* __restrict__ cg,
                                 int P, int total, int ratio,
                                 float* __restrict__ out) {
  const int s = blockIdx.x * 8 + (threadIdx.x >> 5);
  if (s >= total) return;
  const int lane = threadIdx.x & 31;
  int g, idx;
  if (s < P) {
    g = seg_ids[s];
    idx = pos[s];
  } else {
    const int j = s - P;
    g = seg_ids[j / ratio];
    idx = neg[j];
  }
  const float4 x = *(const float4*)(emb + (size_t)idx * NH + (lane << 2));
  const float4 u = *(const float4*)(U + (size_t)g * NH + (lane << 2));
  float p = x.x * u.x + x.y * u.y + x.z * u.z + x.w * u.w;
  for (int o = 16; o; o >>= 1) p += __shfl_xor(p, o, 32);
  if (lane == 0) out[s] = p + cg[g];
}

// ---------------------------------------------------------------------------
// host launcher
// ---------------------------------------------------------------------------
extern "C" void kernel_launch(void* const* d_in, const int* in_sizes, int n_in,
                              void* d_out, int out_size, void* d_ws,
                              size_t ws_size, hipStream_t stream) {
  const float* embedding   = (const float*)d_in[0];
  const float* embedding_  = (const float*)d_in[1];
  const int*   grid_sizes  = (const int*)d_in[2];
  const int*   pos_samples = (const int*)d_in[3];
  const int*   neg_samples = (const int*)d_in[4];
  const float* Wi          = (const float*)d_in[5];
  const float* bi          = (const float*)d_in[6];
  const float* Wk          = (const float*)d_in[7];
  const float* bk          = (const float*)d_in[8];

  const int G  = in_sizes[2];
  const int P  = in_sizes[3];
  const int PN = in_sizes[4];
  const int ratio = PN / P;
  const int total = P + PN;

  // workspace carve-out (256B aligned chunks), ~11.5 MB total
  char* ws = (char*)d_ws;
  auto carve = [&](size_t bytes) -> char* {
    char* p = ws;
    ws += (bytes + 255) & ~(size_t)255;
    return p;
  };
  int*   offsets = (int*)carve((size_t)(G + 1) * sizeof(int));
  int*   seg_ids = (int*)carve((size_t)P * sizeof(int));
  float* M       = (float*)carve((size_t)G * NH * sizeof(float));
  float* T1      = (float*)carve((size_t)NH * NH * sizeof(float));
  float* Amat    = (float*)carve((size_t)NH * NH * sizeof(float));
  float* vecs    = (float*)carve((size_t)(2 * NH + 4) * sizeof(float));
  float* U       = (float*)carve((size_t)G * NH * sizeof(float));
  float* cg      = (float*)carve((size_t)G * sizeof(float));

  td_scan_kernel<<<1, 1024, 0, stream>>>(grid_sizes, G, offsets);
  td_segfill_kernel<<<(G + 255) / 256, 256, 0, stream>>>(offsets, G, seg_ids);
  td_gridmean_kernel<<<(G + 7) / 8, 256, 0, stream>>>(embedding_, pos_samples,
                                                      offsets, G, M);
  td_setup_kernel<<<1, 256, 0, stream>>>(Wi, bi, Wk, bk, T1, Amat, vecs);
  td_u_wmma_kernel<<<(G + 15) / 16, 256, 0, stream>>>(M, Amat, vecs, U, G);
  td_c_kernel<<<(G + 7) / 8, 256, 0, stream>>>(M, vecs, G, cg);
  td_logits_kernel<<<(total + 7) / 8, 256, 0, stream>>>(
      embedding, pos_samples, neg_samples, seg_ids, U, cg, P, total, ratio,
      (float*)d_out);
}